// ModularFieldVortexLayer_22539988369919
// MI455X (gfx1250) — compile-verified
//
#include <hip/hip_runtime.h>
#include <hip/hip_bf16.h>

// ---------------------------------------------------------------------------
// ModularFieldVortexLayer on MI455X (gfx1250)
//   out[b,i] = LeakyReLU(LN_i( sum_j A[i,j] * feats[b,j] @ W[i, class(b,j)] ))
// Strategy: class-routed aggregation in LDS (~300KB of the 320KB WGP LDS),
// dense f32 WMMA (V_WMMA_F32_16X16X4_F32) over K = 9 classes x 256, with the
// weight table repacked once into WMMA-fragment order so every B-frag is one
// coalesced global_load_b64 from the (L2-resident) 21MB table.
// ---------------------------------------------------------------------------

typedef __attribute__((ext_vector_type(2))) float v2f;
typedef __attribute__((ext_vector_type(8))) float v8f;

#define BATCH     16384
#define NNODES    9
#define HIDDEN    256
#define MODULUS   9
#define BT        32            // batch rows per workgroup tile
#define AG_STRIDE 260           // padded row stride (floats): 16B-aligned rows,
                                // conflict-free A-frag reads (bank step 4/row)

#define IDX_ELEMS   (BATCH * NNODES)                       // 147456 ints
#define WPACK_ELEMS (NNODES * MODULUS * HIDDEN * HIDDEN)   // 5,308,416 floats
#define WPACK_OFF   ((size_t)IDX_ELEMS * sizeof(int))      // byte offset in ws

// ---------------------------------------------------------------------------
// Pass 0: repack W[i][r][k][n] (row-major) into per-fragment order:
//   Wpack[ir][kk4][nblk][lane][v]  with  n = nblk*16 + lane%16,
//                                        k = kk4*4 + 2*(lane/16) + v.
// One thread per (ir, kk4, nblk, lane): 2 scattered b32 reads -> 1 b64 write.
// ---------------------------------------------------------------------------
__global__ __launch_bounds__(256) void vortex_packw_kernel(
    const float* __restrict__ W, float* __restrict__ Wpack)
{
    const int t = (int)(blockIdx.x * blockDim.x + threadIdx.x);
    // total = 81 * 64 * 16 * 32 = 2,654,208
    const int lane = t & 31;
    const int nblk = (t >> 5) & 15;
    const int kk4  = (t >> 9) & 63;
    const int ir   = t >> 15;
    if (ir >= NNODES * MODULUS) return;

    const int k0 = kk4 * 4 + 2 * (lane >> 4);
    const int n  = nblk * 16 + (lane & 15);
    const float* src = W + (size_t)ir * HIDDEN * HIDDEN;

    float2 val;
    val.x = src[(size_t)k0 * HIDDEN + n];
    val.y = src[(size_t)(k0 + 1) * HIDDEN + n];

    float* dst = Wpack + (size_t)ir * HIDDEN * HIDDEN
               + ((size_t)kk4 * 16 + nblk) * 64 + lane * 2;
    *(float2*)dst = val;
}

// ---------------------------------------------------------------------------
// Pass 1: digital-root class per (b, node) row. One wave32 per row.
// root = trunc(fmod(|sum_h feats|, 9)); 0 -> 9; idx = root - 1.
// ---------------------------------------------------------------------------
__global__ __launch_bounds__(256) void vortex_class_kernel(
    const float* __restrict__ feats, int* __restrict__ idx, int nrows)
{
    const int warp = (int)((blockIdx.x * blockDim.x + threadIdx.x) >> 5);
    const int lane = (int)(threadIdx.x & 31);
    if (warp >= nrows) return;

    const float* row = feats + (size_t)warp * HIDDEN;
    float s = 0.f;
#pragma unroll
    for (int t = 0; t < HIDDEN / 32; ++t) s += row[lane + 32 * t];
#pragma unroll
    for (int off = 16; off > 0; off >>= 1) s += __shfl_xor(s, off, 32);

    if (lane == 0) {
        int r = (int)fmodf(fabsf(s), 9.0f);
        if (r == 0) r = 9;
        idx[warp] = r - 1;
    }
}

// ---------------------------------------------------------------------------
// Pass 2: one workgroup per (batch-tile of 32, output node i).
//   LDS agg[class][32][260]: class-routed neighbor aggregation.
//   8 waves: each owns a 32-wide N slice; 2 M-tiles x 2 N-tiles of
//   v_wmma_f32_16x16x4_f32 accumulators, K-loop = 9 classes x 256.
//   Epilogue: fused LayerNorm + LeakyReLU via LDS restage + shfl reduce.
// PACKED selects fragment-ordered W (coalesced b64 B-frags) vs row-major W.
// ---------------------------------------------------------------------------
template <bool PACKED>
__global__ __launch_bounds__(256, 1) void vortex_main_kernel(
    const float* __restrict__ feats,   // [B, N, H]
    const float* __restrict__ Wsrc,    // packed or row-major [N, MOD, H, H]
    const float* __restrict__ gamma,   // [N, H]
    const float* __restrict__ beta,    // [N, H]
    const float* __restrict__ adj,     // [N, N]
    const int*   __restrict__ idx,     // [B, N]
    float*       __restrict__ out)     // [B, N, H]
{
    __shared__ float sAgg[MODULUS][BT][AG_STRIDE];   // 299,520 B
    __shared__ int   sIdx[BT][NNODES];               //   1,152 B

    const int bt   = (int)blockIdx.x;
    const int i    = (int)blockIdx.y;
    const int tid  = (int)threadIdx.x;
    const int wave = tid >> 5;
    const int lane = tid & 31;
    const int b0   = bt * BT;

    // In-neighbor mask of node i (adjacency row + self), uniform per block.
    unsigned amask = 0u;
    for (int j = 0; j < NNODES; ++j) {
        if (adj[i * NNODES + j] != 0.f || j == i) amask |= (1u << j);
    }

    // Zero the aggregation buffer.
    for (int t = tid; t < MODULUS * BT * AG_STRIDE; t += 256)
        ((float*)sAgg)[t] = 0.f;

    // Stage the class indices for this tile.
    for (int t = tid; t < BT * NNODES; t += 256)
        ((int*)sIdx)[t] = idx[(size_t)(b0 + t / NNODES) * NNODES + (t % NNODES)];
    __syncthreads();

    // Class-routed aggregation. Each batch row is owned by exactly one wave,
    // so LDS read-modify-writes on agg[c][bb][*] never race across waves.
    for (int bb = wave; bb < BT; bb += 8) {
        const int bg = b0 + bb;
        const int h0 = lane * 8;                       // 8 contiguous floats/lane
        for (int j = 0; j < NNODES; ++j) {
            if (!((amask >> j) & 1u)) continue;
            const int c = sIdx[bb][j];
            const float* src = feats + ((size_t)bg * NNODES + j) * HIDDEN;
            float* dst = &sAgg[c][bb][0];
            float4 x0 = *(const float4*)(src + h0);
            float4 x1 = *(const float4*)(src + h0 + 4);
            float4* d = (float4*)(dst + h0);
            float4 a0 = d[0], a1 = d[1];
            a0.x += x0.x; a0.y += x0.y; a0.z += x0.z; a0.w += x0.w;
            a1.x += x1.x; a1.y += x1.y; a1.z += x1.z; a1.w += x1.w;
            d[0] = a0; d[1] = a1;
        }
    }
    __syncthreads();

    // ---- WMMA main loop -------------------------------------------------
    // Per ISA layouts (32-bit A 16x4 / B 4x16): k = 2*(lane/16) + elem,
    // m/n = lane%16. D (16x16 f32): m = elem + 8*(lane/16), n = lane%16.
    v8f acc00 = {}, acc01 = {}, acc10 = {}, acc11 = {};   // [mtile][ntile]

    const int n0    = wave * 32;          // this wave's N slice
    const int nloc  = lane & 15;
    const int khalf = (lane >> 4) * 2;    // 0 or 2
    const size_t Wbase_i = (size_t)i * MODULUS * HIDDEN * HIDDEN;

    for (int r = 0; r < MODULUS; ++r) {
        const float* pA0 = &sAgg[r][nloc][khalf];        // rows 0..15   (m = lane%16)
        const float* pA1 = &sAgg[r][16 + nloc][khalf];   // rows 16..31
        const float* Wr  = Wsrc + Wbase_i + (size_t)r * HIDDEN * HIDDEN;

        if constexpr (PACKED) {
            // Fragment-ordered: B-frag = one coalesced b64 per lane.
            const float* pW0 = Wr + (size_t)(2 * wave) * 64 + lane * 2;
            const float* pW1 = Wr + (size_t)(2 * wave + 1) * 64 + lane * 2;
#pragma unroll 4
            for (int kk4 = 0; kk4 < HIDDEN / 4; ++kk4) {
                const int kk = kk4 * 4;
                v2f a0 = { pA0[kk], pA0[kk + 1] };
                v2f a1 = { pA1[kk], pA1[kk + 1] };
                v2f bl = *(const v2f*)(pW0 + (size_t)kk4 * 1024);
                v2f br = *(const v2f*)(pW1 + (size_t)kk4 * 1024);

                acc00 = __builtin_amdgcn_wmma_f32_16x16x4_f32(
                    false, a0, false, bl, (short)0, acc00, false, false);
                acc01 = __builtin_amdgcn_wmma_f32_16x16x4_f32(
                    false, a0, false, br, (short)0, acc01, false, false);
                acc10 = __builtin_amdgcn_wmma_f32_16x16x4_f32(
                    false, a1, false, bl, (short)0, acc10, false, false);
                acc11 = __builtin_amdgcn_wmma_f32_16x16x4_f32(
                    false, a1, false, br, (short)0, acc11, false, false);
            }
        } else {
            // Row-major fallback: two strided b32 loads per B-frag.
            const float* pW0 = Wr + (size_t)khalf * HIDDEN + (n0 + nloc);
            const float* pW1 = pW0 + 16;
#pragma unroll 4
            for (int kk = 0; kk < HIDDEN; kk += 4) {
                v2f a0 = { pA0[kk], pA0[kk + 1] };
                v2f a1 = { pA1[kk], pA1[kk + 1] };
                v2f bl = { pW0[(size_t)kk * HIDDEN], pW0[(size_t)kk * HIDDEN + HIDDEN] };
                v2f br = { pW1[(size_t)kk * HIDDEN], pW1[(size_t)kk * HIDDEN + HIDDEN] };

                acc00 = __builtin_amdgcn_wmma_f32_16x16x4_f32(
                    false, a0, false, bl, (short)0, acc00, false, false);
                acc01 = __builtin_amdgcn_wmma_f32_16x16x4_f32(
                    false, a0, false, br, (short)0, acc01, false, false);
                acc10 = __builtin_amdgcn_wmma_f32_16x16x4_f32(
                    false, a1, false, bl, (short)0, acc10, false, false);
                acc11 = __builtin_amdgcn_wmma_f32_16x16x4_f32(
                    false, a1, false, br, (short)0, acc11, false, false);
            }
        }
    }
    __syncthreads();   // everyone done reading sAgg before we reuse it

    // ---- Epilogue: restage tile, LayerNorm, LeakyReLU -------------------
    float* sOut = (float*)sAgg;               // [BT][AG_STRIDE] region reused
    const int mhi = (lane >> 4) * 8;
#pragma unroll
    for (int v = 0; v < 8; ++v) {
        sOut[(size_t)(mhi + v) * AG_STRIDE + n0 + nloc]           = acc00[v];
        sOut[(size_t)(mhi + v) * AG_STRIDE + n0 + 16 + nloc]      = acc01[v];
        sOut[(size_t)(16 + mhi + v) * AG_STRIDE + n0 + nloc]      = acc10[v];
        sOut[(size_t)(16 + mhi + v) * AG_STRIDE + n0 + 16 + nloc] = acc11[v];
    }
    __syncthreads();

    // 8 waves x 4 rows each.
    for (int bb = wave * 4; bb < wave * 4 + 4; ++bb) {
        const float* row = sOut + (size_t)bb * AG_STRIDE;
        float s = 0.f, ss = 0.f;
#pragma unroll
        for (int t = 0; t < HIDDEN / 32; ++t) {
            float x = row[lane + 32 * t];
            s += x; ss += x * x;
        }
#pragma unroll
        for (int off = 16; off > 0; off >>= 1) {
            s  += __shfl_xor(s, off, 32);
            ss += __shfl_xor(ss, off, 32);
        }
        const float mean = s * (1.0f / HIDDEN);
        const float var  = ss * (1.0f / HIDDEN) - mean * mean;
        const float rstd = rsqrtf(var + 1e-5f);

#pragma unroll
        for (int t = 0; t < HIDDEN / 32; ++t) {
            const int h = lane + 32 * t;
            float y = (row[h] - mean) * rstd;
            y = y * gamma[i * HIDDEN + h] + beta[i * HIDDEN + h];
            y = (y >= 0.f) ? y : 0.01f * y;
            out[((size_t)(b0 + bb) * NNODES + i) * HIDDEN + h] = y;
        }
    }
}

// ---------------------------------------------------------------------------
extern "C" void kernel_launch(void* const* d_in, const int* in_sizes, int n_in,
                              void* d_out, int out_size, void* d_ws, size_t ws_size,
                              hipStream_t stream)
{
    (void)in_sizes; (void)n_in; (void)out_size;

    const float* feats = (const float*)d_in[0];   // [B, N, H]
    const float* W     = (const float*)d_in[1];   // [N, MOD, H, H]
    const float* gamma = (const float*)d_in[2];   // [N, H]
    const float* beta  = (const float*)d_in[3];   // [N, H]
    const float* adj   = (const float*)d_in[4];   // [N, N]
    float*       out   = (float*)d_out;           // [B, N, H]

    int*   idxbuf = (int*)d_ws;                                   // [B*N]
    float* Wpack  = (float*)((char*)d_ws + WPACK_OFF);            // packed W

    const bool packed =
        ws_size >= WPACK_OFF + (size_t)WPACK_ELEMS * sizeof(float);

    const int nrows   = BATCH * NNODES;           // 147456 rows
    const int blocks1 = (nrows + 7) / 8;          // 8 waves per 256-thread block
    vortex_class_kernel<<<blocks1, 256, 0, stream>>>(feats, idxbuf, nrows);

    dim3 grid(BATCH / BT, NNODES);
    if (packed) {
        const int total   = NNODES * MODULUS * (HIDDEN / 4) * 16 * 32;
        const int blocks0 = (total + 255) / 256;
        vortex_packw_kernel<<<blocks0, 256, 0, stream>>>(W, Wpack);
        vortex_main_kernel<true><<<grid, 256, 0, stream>>>(
            feats, Wpack, gamma, beta, adj, idxbuf, out);
    } else {
        vortex_main_kernel<false><<<grid, 256, 0, stream>>>(
            feats, W, gamma, beta, adj, idxbuf, out);
    }
}